// EMGMAMBA_3745211482739
// MI455X (gfx1250) — compile-verified
//
#include <hip/hip_runtime.h>
#include <hip/hip_bf16.h>
#include <math.h>

// ---------------------------------------------------------------------------
// Model constants (match reference)
// ---------------------------------------------------------------------------
#define BATCH   2
#define SEQ_T   8192
#define FEATS   64
#define D_INNER 256
#define D_STATE 16
#define D_CONV  4
#define DT_RANK 4
#define BT      (BATCH * SEQ_T)          // 16384 rows for the GEMMs
#define NCHUNK  64                        // scan chunks
#define CLEN    (SEQ_T / NCHUNK)          // 128 steps per chunk

typedef __attribute__((ext_vector_type(2))) float v2f;
typedef __attribute__((ext_vector_type(8))) float v8f;

// ---------------------------------------------------------------------------
// Helpers
// ---------------------------------------------------------------------------
__device__ __forceinline__ int refl(int t, int T) {
    if (t < 0)  t = -t;
    if (t >= T) t = 2 * T - 2 - t;
    return t;
}

__device__ __forceinline__ float leaky(float v) {
    return v > 0.f ? v : 0.2f * v;
}

__device__ __forceinline__ float silu(float v) {
    return v / (1.f + __expf(-v));
}

// ---------------------------------------------------------------------------
// conv_in: (B,1,T) -> (B,64,T), 9 taps, reflect pad 4, leaky
// ---------------------------------------------------------------------------
__global__ void conv_in_leaky_k(const float* __restrict__ x,
                                const float* __restrict__ w,   // (64,1,9)
                                const float* __restrict__ bias,
                                float* __restrict__ out, int T) {
    int idx = blockIdx.x * blockDim.x + threadIdx.x;
    int total = BATCH * FEATS * T;
    if (idx >= total) return;
    int t = idx % T;
    int c = (idx / T) % FEATS;
    int b = idx / (T * FEATS);
    const float* xb = x + (size_t)b * T;
    float acc = bias[c];
#pragma unroll
    for (int k = 0; k < 9; ++k)
        acc += xb[refl(t + k - 4, T)] * w[c * 9 + k];
    out[idx] = leaky(acc);
}

// ---------------------------------------------------------------------------
// HNF multiscale filter bank: 64 -> 4x16 channels (taps 3/5/9/15), reflect pad
// ---------------------------------------------------------------------------
__global__ void multi_filter_k(const float* __restrict__ in,
                               const float* __restrict__ w3, const float* __restrict__ b3,
                               const float* __restrict__ w5, const float* __restrict__ b5,
                               const float* __restrict__ w9, const float* __restrict__ b9,
                               const float* __restrict__ w15, const float* __restrict__ b15,
                               float* __restrict__ out, int T) {
    int idx = blockIdx.x * blockDim.x + threadIdx.x;
    int total = BATCH * FEATS * T;
    if (idx >= total) return;
    int t  = idx % T;
    int oc = (idx / T) % FEATS;
    int b  = idx / (T * FEATS);
    int f   = oc >> 4;
    int sub = oc & 15;
    const float* wsel; const float* bsel; int taps;
    if (f == 0)      { wsel = w3;  bsel = b3;  taps = 3;  }
    else if (f == 1) { wsel = w5;  bsel = b5;  taps = 5;  }
    else if (f == 2) { wsel = w9;  bsel = b9;  taps = 9;  }
    else             { wsel = w15; bsel = b15; taps = 15; }
    int pad = (taps - 1) >> 1;
    const float* ib = in + (size_t)b * FEATS * T;
    float acc = bsel[sub];
    for (int ic = 0; ic < FEATS; ++ic) {
        const float* row = ib + (size_t)ic * T;
        const float* wr  = wsel + (size_t)(sub * FEATS + ic) * taps;
        for (int k = 0; k < taps; ++k)
            acc += row[refl(t + k - pad, T)] * wr[k];
    }
    out[idx] = acc;
}

// ---------------------------------------------------------------------------
// Generic 64-in-channel 9-tap reflect conv.  mode bit0: leaky, bit1: +residual.
// Each 256-thread block covers one (b,oc) t-range, so the 576-float weight
// slab for that oc is staged into LDS with CDNA5 async global->LDS loads.
// ---------------------------------------------------------------------------
__global__ void conv9_k(const float* __restrict__ in,
                        const float* __restrict__ w,    // (OC,64,9)
                        const float* __restrict__ bias,
                        float* __restrict__ out,
                        const float* __restrict__ resid,
                        int OC, int T, int mode) {
    __shared__ float wsh[FEATS * 9];      // 576 floats = 2304 B
    int idx      = blockIdx.x * blockDim.x + threadIdx.x;
    int base_idx = blockIdx.x * blockDim.x;
    int oc = (base_idx / T) % OC;         // uniform across block (256 | T)
    int b  = base_idx / (T * OC);         // uniform across block
    const float* wp = w + (size_t)oc * FEATS * 9;

    if (threadIdx.x < (FEATS * 9) / 4) {  // 144 lanes x b128 = 2304 B
        unsigned lds_off = (unsigned)(unsigned long long)(uintptr_t)(&wsh[threadIdx.x * 4]);
        unsigned long long ga = (unsigned long long)(uintptr_t)(wp + threadIdx.x * 4);
        asm volatile("global_load_async_to_lds_b128 %0, %1, off"
                     :: "v"(lds_off), "v"(ga)
                     : "memory");
    }
    asm volatile("s_wait_asynccnt 0" ::: "memory");
    __syncthreads();

    int t = idx % T;
    const float* ib = in + (size_t)b * FEATS * T;
    float acc = bias[oc];
    for (int ic = 0; ic < FEATS; ++ic) {
        const float* row = ib + (size_t)ic * T;
        const float* wr  = &wsh[ic * 9];
#pragma unroll
        for (int k = 0; k < 9; ++k)
            acc += row[refl(t + k - 4, T)] * wr[k];
    }
    if (mode & 1) acc = leaky(acc);
    if (mode & 2) acc += resid[idx];
    out[idx] = acc;
}

// ---------------------------------------------------------------------------
// Instance norm (channels 0..31) + leaky over all 64 channels.
// ---------------------------------------------------------------------------
__global__ void inorm_leaky_k(const float* __restrict__ in,
                              float* __restrict__ out, int T) {
    int bc = blockIdx.x;
    int c  = bc & 63;
    const float* src = in + (size_t)bc * T;
    float* dst = out + (size_t)bc * T;
    if (c < 32) {
        __shared__ float red_s[8];
        __shared__ float red_q[8];
        float s = 0.f, q = 0.f;
        for (int t = threadIdx.x; t < T; t += blockDim.x) {
            float v = src[t]; s += v; q += v * v;
        }
#pragma unroll
        for (int off = 16; off >= 1; off >>= 1) {
            s += __shfl_xor(s, off, 32);
            q += __shfl_xor(q, off, 32);
        }
        int wid = threadIdx.x >> 5, lane = threadIdx.x & 31;
        if (lane == 0) { red_s[wid] = s; red_q[wid] = q; }
        __syncthreads();
        if (threadIdx.x == 0) {
            float S = 0.f, Q = 0.f;
            int nw = blockDim.x >> 5;
            for (int i = 0; i < nw; ++i) { S += red_s[i]; Q += red_q[i]; }
            red_s[0] = S; red_q[0] = Q;
        }
        __syncthreads();
        float invT = 1.f / (float)T;
        float mu   = red_s[0] * invT;
        float var  = red_q[0] * invT - mu * mu;
        float inv  = rsqrtf(var + 1e-5f);
        for (int t = threadIdx.x; t < T; t += blockDim.x)
            dst[t] = leaky((src[t] - mu) * inv);
    } else {
        for (int t = threadIdx.x; t < T; t += blockDim.x)
            dst[t] = leaky(src[t]);
    }
}

// ---------------------------------------------------------------------------
// RMS norm + transpose: (B,64,T) -> U (B*T, 64). One wave per (b,t) row.
// ---------------------------------------------------------------------------
__global__ void rmsnorm_transpose_k(const float* __restrict__ hin,
                                    const float* __restrict__ w,
                                    float* __restrict__ U, int T) {
    int wave = blockIdx.x * (blockDim.x >> 5) + (threadIdx.x >> 5);
    int lane = threadIdx.x & 31;
    if (wave >= BATCH * T) return;
    int b = wave / T, t = wave % T;
    const float* base = hin + (size_t)b * FEATS * T + t;
    float v0 = base[(size_t)lane * T];
    float v1 = base[(size_t)(lane + 32) * T];
    float ss = v0 * v0 + v1 * v1;
#pragma unroll
    for (int off = 16; off >= 1; off >>= 1)
        ss += __shfl_xor(ss, off, 32);
    float r = rsqrtf(ss * (1.f / 64.f) + 1e-5f);
    float* o = U + (size_t)wave * 64;
    o[lane]      = v0 * r * w[lane];
    o[lane + 32] = v1 * r * w[lane + 32];
}

// ---------------------------------------------------------------------------
// fp32 GEMM via V_WMMA_F32_16X16X4_F32, branch-free inner loop.
// Y[m,n] = sum_k A[m,k] * W[n,k].  A: (M,K) rm, W: (N,K) rm, Y: (M,Nstride).
// Invalid columns (n >= Nvalid) read W row 0 (in-bounds) and write garbage
// into padded scratch columns that are never consumed.
// ---------------------------------------------------------------------------
__global__ void gemm_wmma_f32_k(const float* __restrict__ A,
                                const float* __restrict__ W,
                                float* __restrict__ Y,
                                int M, int K, int Nstride, int Nvalid,
                                int tilesN, int totTiles) {
    int wave = blockIdx.x * (blockDim.x >> 5) + (threadIdx.x >> 5);
    int lane = threadIdx.x & 31;
    if (wave >= totTiles) return;            // uniform per-wave
    int mt = wave / tilesN;
    int nt = wave % tilesN;
    int lrow = lane & 15;
    int half = lane >> 4;
    int arow = mt * 16 + lrow;
    int ncol = nt * 16 + lrow;
    int nclamp = (ncol < Nvalid) ? ncol : 0;  // keep loads in-bounds
    const float* Ap = A + (size_t)arow * K + 2 * half;
    const float* Wp = W + (size_t)nclamp * K + 2 * half;

    v8f c = {0.f, 0.f, 0.f, 0.f, 0.f, 0.f, 0.f, 0.f};
    for (int k0 = 0; k0 < K; k0 += 4) {
        v2f a   = *(const v2f*)(Ap + k0);
        v2f bfr = *(const v2f*)(Wp + k0);
        __builtin_prefetch(Wp + k0 + 32, 0, 1);   // global_prefetch_b8
        c = __builtin_amdgcn_wmma_f32_16x16x4_f32(
                /*neg_a=*/false, a, /*neg_b=*/false, bfr,
                /*c_mod=*/(short)0, c, /*reuse_a=*/false, /*reuse_b=*/false);
    }
    float* Yp = Y + (size_t)(mt * 16 + 8 * half) * Nstride + nt * 16 + lrow;
#pragma unroll
    for (int v = 0; v < 8; ++v)
        Yp[(size_t)v * Nstride] = c[v];
}

// ---------------------------------------------------------------------------
// Depthwise causal conv (D_CONV=4) over x half of xz, + silu.
// ---------------------------------------------------------------------------
__global__ void dwconv_silu_k(const float* __restrict__ xz,
                              const float* __restrict__ cw,   // (256,1,4)
                              const float* __restrict__ cb,
                              float* __restrict__ xc, int T) {
    int idx = blockIdx.x * blockDim.x + threadIdx.x;
    if (idx >= BT * D_INNER) return;
    int d  = idx & (D_INNER - 1);
    int bt = idx >> 8;
    int b  = bt / T;
    int t  = bt % T;
    float acc = cb[d];
#pragma unroll
    for (int k = 0; k < D_CONV; ++k) {
        int tt = t - (D_CONV - 1) + k;
        if (tt >= 0)
            acc += xz[(size_t)(b * T + tt) * 512 + d] * cw[d * D_CONV + k];
    }
    xc[(size_t)bt * D_INNER + d] = silu(acc);
}

// ---------------------------------------------------------------------------
// dt = softplus(x_dbl[:, :4] @ dt_proj_w^T + dt_proj_b)
// ---------------------------------------------------------------------------
__global__ void dt_softplus_k(const float* __restrict__ xdbl,
                              const float* __restrict__ dw,   // (256,4)
                              const float* __restrict__ db,
                              float* __restrict__ dt, int T) {
    int idx = blockIdx.x * blockDim.x + threadIdx.x;
    if (idx >= BT * D_INNER) return;
    int d  = idx & (D_INNER - 1);
    int bt = idx >> 8;
    const float* xd = xdbl + (size_t)bt * 48;
    float s = db[d];
#pragma unroll
    for (int r = 0; r < DT_RANK; ++r)
        s += xd[r] * dw[d * DT_RANK + r];
    float sp = (s > 20.f) ? s : log1pf(__expf(s));
    dt[(size_t)bt * D_INNER + d] = sp;
}

// ---------------------------------------------------------------------------
// Chunked parallel scan.  h_t = dA_t*h_{t-1} + dBx_t is affine, so:
//   pass A: per chunk, P = prod(dA), S = chunk scan with h0 = 0
//   pass B: serial prefix over the 64 chunks per (b,d,n): h_start chain
//   pass C: re-run chunk from correct h_start, emit gated outputs.
// P/S/Hst layout: [((b*256+d)*16+n)*NCHUNK + c]
// ---------------------------------------------------------------------------
__global__ void scan_chunkA_k(const float* __restrict__ xdbl,
                              const float* __restrict__ dt,
                              const float* __restrict__ xconv,
                              const float* __restrict__ A_log,
                              float* __restrict__ P,
                              float* __restrict__ Sv, int T) {
    int idx = blockIdx.x * blockDim.x + threadIdx.x;
    if (idx >= BATCH * D_INNER * NCHUNK) return;
    int c = idx % NCHUNK;
    int d = (idx / NCHUNK) % D_INNER;
    int b = idx / (NCHUNK * D_INNER);
    float a[D_STATE], h[D_STATE], p[D_STATE];
#pragma unroll
    for (int n = 0; n < D_STATE; ++n) {
        a[n] = -__expf(A_log[d * D_STATE + n]);
        h[n] = 0.f; p[n] = 1.f;
    }
    int t0 = c * CLEN;
    for (int t = t0; t < t0 + CLEN; ++t) {
        size_t bt = (size_t)b * T + t;
        float dtv = dt[bt * D_INNER + d];
        float xv  = xconv[bt * D_INNER + d];
        const float* xd = xdbl + bt * 48;
#pragma unroll
        for (int n = 0; n < D_STATE; ++n) {
            float dA = __expf(dtv * a[n]);
            p[n] *= dA;
            h[n] = dA * h[n] + dtv * xd[DT_RANK + n] * xv;
        }
    }
    size_t base = ((size_t)(b * D_INNER + d) * D_STATE) * NCHUNK + c;
#pragma unroll
    for (int n = 0; n < D_STATE; ++n) {
        P [base + (size_t)n * NCHUNK] = p[n];
        Sv[base + (size_t)n * NCHUNK] = h[n];
    }
}

__global__ void scan_prefix_k(const float* __restrict__ P,
                              const float* __restrict__ Sv,
                              float* __restrict__ Hst) {
    int idx = blockIdx.x * blockDim.x + threadIdx.x;
    if (idx >= BATCH * D_INNER * D_STATE) return;
    size_t base = (size_t)idx * NCHUNK;
    float hs = 0.f;
    for (int c = 0; c < NCHUNK; ++c) {
        Hst[base + c] = hs;
        hs = P[base + c] * hs + Sv[base + c];
    }
}

__global__ void scan_chunkC_k(const float* __restrict__ xdbl,
                              const float* __restrict__ dt,
                              const float* __restrict__ xconv,
                              const float* __restrict__ xz,
                              const float* __restrict__ A_log,
                              const float* __restrict__ Dp,
                              const float* __restrict__ Hst,
                              float* __restrict__ y, int T) {
    int idx = blockIdx.x * blockDim.x + threadIdx.x;
    if (idx >= BATCH * D_INNER * NCHUNK) return;
    int c = idx % NCHUNK;
    int d = (idx / NCHUNK) % D_INNER;
    int b = idx / (NCHUNK * D_INNER);
    size_t base = ((size_t)(b * D_INNER + d) * D_STATE) * NCHUNK + c;
    float a[D_STATE], h[D_STATE];
#pragma unroll
    for (int n = 0; n < D_STATE; ++n) {
        a[n] = -__expf(A_log[d * D_STATE + n]);
        h[n] = Hst[base + (size_t)n * NCHUNK];
    }
    float Dd = Dp[d];
    int t0 = c * CLEN;
    for (int t = t0; t < t0 + CLEN; ++t) {
        size_t bt = (size_t)b * T + t;
        float dtv = dt[bt * D_INNER + d];
        float xv  = xconv[bt * D_INNER + d];
        const float* xd = xdbl + bt * 48;
        float acc = 0.f;
#pragma unroll
        for (int n = 0; n < D_STATE; ++n) {
            float dA = __expf(dtv * a[n]);
            h[n] = dA * h[n] + dtv * xd[DT_RANK + n] * xv;
            acc += h[n] * xd[DT_RANK + D_STATE + n];
        }
        float zv = xz[bt * 512 + D_INNER + d];
        y[bt * D_INNER + d] = (acc + Dd * xv) * silu(zv);
    }
}

// ---------------------------------------------------------------------------
// mamba out (BT,64) + residual (B,64,T) -> (B,64,T)
// ---------------------------------------------------------------------------
__global__ void add_res_transpose_k(const float* __restrict__ mout,
                                    const float* __restrict__ res,
                                    float* __restrict__ out, int T) {
    int idx = blockIdx.x * blockDim.x + threadIdx.x;
    int total = BATCH * FEATS * T;
    if (idx >= total) return;
    int t = idx % T;
    int c = (idx / T) % FEATS;
    int b = idx / (T * FEATS);
    out[idx] = mout[(size_t)(b * T + t) * FEATS + c] + res[idx];
}

// ---------------------------------------------------------------------------
// Host-side launcher
// ---------------------------------------------------------------------------
extern "C" void kernel_launch(void* const* d_in, const int* in_sizes, int n_in,
                              void* d_out, int out_size, void* d_ws, size_t ws_size,
                              hipStream_t stream) {
    (void)in_sizes; (void)out_size; (void)ws_size;
    const float* in[64];
    for (int i = 0; i < n_in && i < 64; ++i) in[i] = (const float*)d_in[i];

    const int T = SEQ_T;
    float* ws = (float*)d_ws;
    const size_t S = (size_t)BATCH * FEATS * T;   // 1,048,576 floats (4 MB)

    float* buf0  = ws;             // activations (B,64,T)
    float* buf1  = ws + 1 * S;
    float* buf2  = ws + 2 * S;     // hnf scratch
    float* buf3  = ws + 3 * S;     // hnf scratch
    float* U     = ws + 4 * S;     // (BT,64)
    float* xz    = ws + 5 * S;     // (BT,512)  -> spans 8 S
    float* xconv = ws + 13 * S;    // (BT,256)  -> spans 4 S
    float* xdbl  = ws + 17 * S;    // (BT,48)   -> < 1 S
    float* dtb   = ws + 18 * S;    // (BT,256)  -> spans 4 S
    float* ybuf  = ws + 22 * S;    // (BT,256)  -> spans 4 S
    float* mout  = ws + 26 * S;    // (BT,64)
    float* scanP = ws + 27 * S;    // (B*256*16*64) = S/2
    float* scanS = scanP + (S / 2);
    float* scanH = scanS + (S / 2);        // ends at 28.5 * S (~114 MB)

    const int ELEM_BLK = 256;
    const int nBCT = (int)(S / ELEM_BLK);          // 4096 blocks for B*64*T grids

    // Input index map (setup_inputs insertion order):
    // 0:x 1:conv_in_w 2:conv_in_b 3..14:hnf_enc 15..24:mamba 25..36:hnf_dec
    // 37:conv_out_w 38:conv_out_b
    auto run_hnf = [&](const float* cur, float* outb, int p) {
        multi_filter_k<<<nBCT, ELEM_BLK, 0, stream>>>(
            cur, in[p + 0], in[p + 1], in[p + 2], in[p + 3],
            in[p + 4], in[p + 5], in[p + 6], in[p + 7], buf2, T);
        conv9_k<<<nBCT, ELEM_BLK, 0, stream>>>(
            buf2, in[p + 8], in[p + 9], buf3, nullptr, 64, T, /*mode=*/0);
        inorm_leaky_k<<<BATCH * FEATS, 256, 0, stream>>>(buf3, buf2, T);
        conv9_k<<<nBCT, ELEM_BLK, 0, stream>>>(
            buf2, in[p + 10], in[p + 11], outb, cur, 64, T, /*mode=*/3);
    };

    // ---- conv_in + leaky -> buf0
    conv_in_leaky_k<<<nBCT, ELEM_BLK, 0, stream>>>(in[0], in[1], in[2], buf0, T);

    // ---- hnf_enc: buf0 -> buf1
    run_hnf(buf0, buf1, 3);

    // ---- mamba block on buf1 -> buf0
    const int m = 15;
    rmsnorm_transpose_k<<<(BT + 7) / 8, 256, 0, stream>>>(buf1, in[m + 0], U, T);

    // in_proj: (BT,64) x (512,64)^T -> xz (BT,512)
    {
        int tilesN = 512 / 16, tot = (BT / 16) * tilesN;     // 32768 tiles
        gemm_wmma_f32_k<<<(tot + 3) / 4, 128, 0, stream>>>(
            U, in[m + 1], xz, BT, 64, 512, 512, tilesN, tot);
    }
    // depthwise causal conv + silu -> xconv
    dwconv_silu_k<<<(BT * D_INNER) / 256, 256, 0, stream>>>(
        xz, in[m + 2], in[m + 3], xconv, T);
    // x_proj: (BT,256) x (36,256)^T -> xdbl (BT,48 padded, 36 valid)
    {
        int tilesN = 3, tot = (BT / 16) * tilesN;            // 3072 tiles
        gemm_wmma_f32_k<<<(tot + 3) / 4, 128, 0, stream>>>(
            xconv, in[m + 4], xdbl, BT, 256, 48, 36, tilesN, tot);
    }
    // dt = softplus(...)
    dt_softplus_k<<<(BT * D_INNER) / 256, 256, 0, stream>>>(
        xdbl, in[m + 5], in[m + 6], dtb, T);
    // chunked parallel selective scan -> ybuf
    scan_chunkA_k<<<(BATCH * D_INNER * NCHUNK) / 256, 256, 0, stream>>>(
        xdbl, dtb, xconv, in[m + 7], scanP, scanS, T);
    scan_prefix_k<<<(BATCH * D_INNER * D_STATE) / 256, 256, 0, stream>>>(
        scanP, scanS, scanH);
    scan_chunkC_k<<<(BATCH * D_INNER * NCHUNK) / 256, 256, 0, stream>>>(
        xdbl, dtb, xconv, xz, in[m + 7], in[m + 8], scanH, ybuf, T);
    // out_proj: (BT,256) x (64,256)^T -> mout (BT,64)
    {
        int tilesN = 4, tot = (BT / 16) * tilesN;            // 4096 tiles
        gemm_wmma_f32_k<<<(tot + 3) / 4, 128, 0, stream>>>(
            ybuf, in[m + 9], mout, BT, 256, 64, 64, tilesN, tot);
    }
    // residual add + transpose back -> buf0
    add_res_transpose_k<<<nBCT, ELEM_BLK, 0, stream>>>(mout, buf1, buf0, T);

    // ---- hnf_dec: buf0 -> buf1
    run_hnf(buf0, buf1, 25);

    // ---- conv_out: (B,64,T) -> (B,1,T) straight into d_out
    conv9_k<<<(BATCH * 1 * T) / ELEM_BLK, ELEM_BLK, 0, stream>>>(
        buf1, in[37], in[38], (float*)d_out, nullptr, 1, T, /*mode=*/0);
}